// WaveletLinearLayer_36086315221704
// MI455X (gfx1250) — compile-verified
//
#include <hip/hip_runtime.h>

typedef __attribute__((ext_vector_type(2))) float v2f;
typedef __attribute__((ext_vector_type(4))) float f4v;
typedef __attribute__((ext_vector_type(8))) float v8f;

#define IN_DIM   262144   // 4*256*256
#define OUT_DIM  64
#define BATCH    256
#define KSPLIT   32       // 512 (ch,i) pairs / 16 per block
#define PAIRS_PER_KS 16
#define XSTR  260         // 256 + 4 pad  (stride mod 64 banks = 4 -> conflict free)
#define WSTR2 68          // 64 + 4 pad
#define LX_FLOATS (16 * XSTR)        // 4160
#define LW_FLOATS (256 * WSTR2)      // 17408 : [f 0..3][o 0..63][j 0..63]
#define SMEM_BYTES ((LX_FLOATS + LW_FLOATS) * 4)   // 86272 B dynamic LDS

#define WMMA_F32(ACC, A, Bf) \
  (ACC) = __builtin_amdgcn_wmma_f32_16x16x4_f32(false, (A), false, (Bf), \
                                                (short)0, (ACC), false, false)

// One sub-band's 4 N-tile WMMAs fed by one A fragment.
#define DO_SUBBAND(AFRAG, FIDX) do {                                  \
    const float* wj_ = lw + ((FIDX) * 64 + nrow) * WSTR2 + jA;        \
    v2f b0_ = *(const v2f*)(wj_);                                     \
    v2f b1_ = *(const v2f*)(wj_ + 16 * WSTR2);                        \
    v2f b2_ = *(const v2f*)(wj_ + 32 * WSTR2);                        \
    v2f b3_ = *(const v2f*)(wj_ + 48 * WSTR2);                        \
    WMMA_F32(acc0, AFRAG, b0_);                                       \
    WMMA_F32(acc1, AFRAG, b1_);                                       \
    WMMA_F32(acc2, AFRAG, b2_);                                       \
    WMMA_F32(acc3, AFRAG, b3_);                                       \
  } while (0)

// Phase 1: fused Haar + split-K GEMM, one block = 16 batch rows x ALL 64 outs.
// grid = (16 mtiles, 32 ksplits), block = 128 threads = 4 waves.
// Wave w owns j-quarter [w*16, w*16+16) and computes ALL four sub-bands with
// compile-time Haar signs: p/m/q/r shared adds -> 4 A fragments -> 16 WMMAs.
__global__ void __launch_bounds__(128)
wavelet_wmma_gemm(const float* __restrict__ x, const float* __restrict__ W,
                  float* __restrict__ ws) {
  extern __shared__ float smem[];
  float* lx = smem;               // [16 rows][XSTR]: cols 0..127 even img row, 128..255 odd
  float* lw = smem + LX_FLOATS;   // [(f*64+o)][WSTR2], pre-scaled by 0.5

  const int mtile = blockIdx.x;          // 0..15
  const int ks    = blockIdx.y;          // 0..31
  const int tid   = threadIdx.x;
  const int wave  = tid >> 5;            // 0..3 == j-quarter
  const int lane  = tid & 31;
  const int nrow  = lane & 15;           // A: M index / B: N index
  const int hi    = lane >> 4;           // K half selector (K = 2*hi, 2*hi+1)

  v8f acc0 = {}, acc1 = {}, acc2 = {}, acc3 = {};

  for (int p = 0; p < PAIRS_PER_KS; ++p) {
    const int pair = ks * PAIRS_PER_KS + p;   // 0..511
    const int ch = pair >> 7;                 // channel 0..3
    const int ii = pair & 127;                // image row-pair index

    for (int half = 0; half < 2; ++half) {
      __syncthreads();
      // ---- stage x: 16 batch rows x 256 floats (128 even-row + 128 odd-row)
      {
        const int r  = tid >> 3;   // 0..15
        const int c0 = tid & 7;    // float4 lane
        const float* srcBase = x + (size_t)(mtile * 16 + r) * IN_DIM
                                 + (size_t)ch * 65536 + (size_t)ii * 512
                                 + half * 128;
        float* dst = lx + r * XSTR;
        #pragma unroll
        for (int t = 0; t < 8; ++t) {
          int j4 = c0 + t * 8;                 // 0..63 float4 slots
          const float* src = (j4 < 32) ? (srcBase + 4 * j4)
                                       : (srcBase + 256 + 4 * (j4 - 32));
          *(f4v*)(dst + 4 * j4) = *(const f4v*)src;
        }
        if (half == 0 && p + 1 < PAIRS_PER_KS)
          __builtin_prefetch(srcBase + 512, 0, 0);   // next pair's x segment
      }
      // ---- stage W: 256 (f,o) rows x 64 floats, pre-scaled by 0.5 (Haar factor)
      {
        const int rbase = tid >> 4;          // 0..7
        const int c     = tid & 15;          // float4 slot within row
        #pragma unroll
        for (int t = 0; t < 32; ++t) {
          const int row = rbase + t * 8;     // 0..255
          const int fw = row >> 6, ow = row & 63;
          const float* src = W + (size_t)ow * IN_DIM
                               + (size_t)(ch * 4 + fw) * 16384
                               + (size_t)ii * 128 + half * 64 + 4 * c;
          f4v v = *(const f4v*)src;
          v.x *= 0.5f; v.y *= 0.5f; v.z *= 0.5f; v.w *= 0.5f;
          *(f4v*)(lw + row * WSTR2 + 4 * c) = v;
        }
      }
      __syncthreads();

      // ---- this wave's 4 K-steps; each butterfly feeds 16 WMMAs (4 f x 4 nt)
      const float* xrow = lx + nrow * XSTR;
      const int jq = wave * 16;
      #pragma unroll 2
      for (int j0 = jq; j0 < jq + 16; j0 += 4) {
        const int jA = j0 + 2 * hi;          // this lane's local K base
        f4v top = *(const f4v*)(xrow + 2 * jA);        // a0,b0,a1,b1
        f4v bot = *(const f4v*)(xrow + 128 + 2 * jA);  // c0,d0,c1,d1
        // Shared Haar butterfly (constant signs -> pure add/sub, no muls)
        const float p0 = top.x + top.y, m0 = top.x - top.y;
        const float q0 = bot.x + bot.y, r0 = bot.x - bot.y;
        const float p1 = top.z + top.w, m1 = top.z - top.w;
        const float q1 = bot.z + bot.w, r1 = bot.z - bot.w;
        v2f aLL, aLH, aHL, aHH;
        aLL.x = p0 + q0; aLL.y = p1 + q1;
        aLH.x = p0 - q0; aLH.y = p1 - q1;
        aHL.x = m0 + r0; aHL.y = m1 + r1;
        aHH.x = m0 - r0; aHH.y = m1 - r1;
        DO_SUBBAND(aLL, 0);
        DO_SUBBAND(aLH, 1);
        DO_SUBBAND(aHL, 2);
        DO_SUBBAND(aHH, 3);
      }
    }
  }

  // ---- deterministic cross-wave reduce: 4 waves x 4 ntiles x 256 elems
  __syncthreads();
  float* red = smem;   // [wave][nt stride 260][elem], 4*1040 = 4160 floats
  #pragma unroll
  for (int v = 0; v < 8; ++v) {
    const int elem = (v + hi * 8) * 16 + nrow;   // C/D layout: M=v+8*hi, N=lane&15
    red[wave * 1040 + 0 * 260 + elem] = acc0[v];
    red[wave * 1040 + 1 * 260 + elem] = acc1[v];
    red[wave * 1040 + 2 * 260 + elem] = acc2[v];
    red[wave * 1040 + 3 * 260 + elem] = acc3[v];
  }
  __syncthreads();
  for (int e = tid; e < 1024; e += 128) {
    const int nt = e >> 8, elem = e & 255;
    const int idx = nt * 260 + elem;
    float s = red[idx] + red[1040 + idx] + red[2080 + idx] + red[3120 + idx];
    ws[((size_t)ks * 16 + mtile) * 1024 + e] = s;
  }
}

// Phase 2: deterministic split-K reduction + bias. grid = 64 tiles, 256 thr.
__global__ void __launch_bounds__(256)
splitk_reduce_bias(const float* __restrict__ ws, const float* __restrict__ bias,
                   float* __restrict__ out) {
  const int tile = blockIdx.x;     // 0..63  (mtile*4 + nt)
  const int e    = threadIdx.x;    // 0..255 tile element (m*16+n)
  const int mtile = tile >> 2, nt = tile & 3;
  const int m = e >> 4, n = e & 15;
  const int b = mtile * 16 + m;
  const int o = nt * 16 + n;
  float s = 0.f;
  #pragma unroll
  for (int k = 0; k < KSPLIT; ++k)
    s += ws[((size_t)k * 16 + mtile) * 1024 + nt * 256 + e];
  out[(size_t)b * OUT_DIM + o] = s + bias[o];
}

extern "C" void kernel_launch(void* const* d_in, const int* in_sizes, int n_in,
                              void* d_out, int out_size, void* d_ws, size_t ws_size,
                              hipStream_t stream) {
  (void)in_sizes; (void)n_in; (void)out_size; (void)ws_size;
  const float* x = (const float*)d_in[0];   // (256, 262144) f32
  const float* W = (const float*)d_in[1];   // (64, 262144) f32
  const float* b = (const float*)d_in[2];   // (64,) f32
  float* out = (float*)d_out;               // (256, 64) f32
  float* ws  = (float*)d_ws;                // 32*16*1024 f32 partials = 2 MB

  wavelet_wmma_gemm<<<dim3(16, KSPLIT), 128, SMEM_BYTES, stream>>>(x, W, ws);
  splitk_reduce_bias<<<64, 256, 0, stream>>>(ws, b, out);
}